// GraphAttentionLayer_51049981280356
// MI455X (gfx1250) — compile-verified
//
#include <hip/hip_runtime.h>
#include <hip/hip_bf16.h>
#include <math.h>

#define N_NODES   50000
#define N_EDGES   800000
#define D_MODEL   64
#define NUM_HEADS 8
#define D_HEAD    8
#define EDGE_IN   16
#define LN_EPS    1e-5f
#define SMAX_EPS  1e-16f

typedef __attribute__((ext_vector_type(2))) float    v2f;
typedef __attribute__((ext_vector_type(8))) float    v8f;
typedef __attribute__((ext_vector_type(4))) unsigned v4u;
typedef __attribute__((ext_vector_type(8))) int      v8i;
typedef __attribute__((ext_vector_type(4))) int      v4i;

#if __has_builtin(__builtin_amdgcn_tensor_load_to_lds) && \
    __has_builtin(__builtin_amdgcn_s_wait_tensorcnt)
#define HAVE_TDM 1
#else
#define HAVE_TDM 0
#endif

// ---- monotonic float<->uint mapping for atomicMax on signed floats ----
__device__ __forceinline__ unsigned f2ord(float f) {
    unsigned u = __float_as_uint(f);
    return (u & 0x80000000u) ? ~u : (u | 0x80000000u);
}
__device__ __forceinline__ float ord2f(unsigned u) {
    return __uint_as_float((u & 0x80000000u) ? (u & 0x7fffffffu) : ~u);
}
#define ORD_NEG_INF 0x007FFFFFu   // f2ord(-inf)

#if HAVE_TDM
// TDM: DMA a 64x64 f32 row-major matrix from global into LDS at lds_off.
// D# per CDNA5 ISA 8.3/8.4: group0 = {flags, lds_addr, global_addr lo/hi+type},
// group1 = {data_size=4B, tensor 64x64 stride 64, tile 64x64}.
// This toolchain exposes the 6-arg builtin:
//   (uint32x4 g0, int32x8 g1, int32x4 g2, int32x4 g3, int32x8 g4, i32 cpol)
__device__ __forceinline__ void tdm_load_mat64x64(const float* gptr,
                                                  unsigned lds_off) {
    unsigned long long ga = (unsigned long long)gptr;
    v4u g0;
    g0.x = 0x1u;                                        // count=1, no gather
    g0.y = lds_off;                                     // lds_addr (bytes)
    g0.z = (unsigned)(ga & 0xFFFFFFFFull);              // global_addr[31:0]
    g0.w = (unsigned)((ga >> 32) & 0x01FFFFFFull)       // global_addr[56:32]
           | 0x80000000u;                               // type=2 ("image")
    v8i g1;
    g1[0] = 0x00020000;        // workgroup_mask=0, data_size=2 (4 bytes)
    g1[1] = (64 << 16);        // tensor_dim0 = 64            (bit 48)
    g1[2] = (64 << 16);        // tensor_dim1 = 64            (bit 80)
    g1[3] = (64 << 16);        // tile_dim0   = 64            (bit 112)
    g1[4] = 64;                // tile_dim1   = 64            (bit 128)
    g1[5] = 64;                // tensor_dim0_stride = 64     (bit 160)
    g1[6] = (4096 << 16);      // tensor_dim1_stride = 4096   (bit 208)
    g1[7] = 0;
    v4i z4 = {0, 0, 0, 0};                 // groups 2/3 unused for 2D tile
    v8i z8 = {0, 0, 0, 0, 0, 0, 0, 0};     // extra group (clang-23 form)
    __builtin_amdgcn_tensor_load_to_lds(g0, g1, z4, z4, z8, 0);
}
#endif

// ====================================================================
// Kernel 0: init per-node accumulators (agg=0, denom=0, smax=-inf)
// ====================================================================
__global__ __launch_bounds__(256) void init_kernel(float* __restrict__ agg,
                                                   float* __restrict__ denom,
                                                   unsigned* __restrict__ smax) {
    int i = blockIdx.x * 256 + threadIdx.x;
    if (i < N_NODES * D_MODEL) agg[i] = 0.0f;
    if (i < N_NODES * NUM_HEADS) { denom[i] = 0.0f; smax[i] = ORD_NEG_INF; }
}

// ====================================================================
// Kernel 1: fused Q/K/V projection with V_WMMA_F32_16X16X4_F32.
// One block = 16 rows of x; 4 waves, wave w owns output columns
// [16w,16w+16) for all three matrices. Weights staged into LDS by the
// Tensor Data Mover (wave 0 issues 3 descriptors, s_wait_tensorcnt 0);
// the 16x64 x tile is staged via coalesced float4 loads. A fragments
// are loaded once into registers and reused across q/k/v.
// N_NODES = 50000 = 3125 * 16 exactly, no tail.
// ====================================================================
__global__ __launch_bounds__(128) void qkv_wmma_kernel(
    const float* __restrict__ x,
    const float* __restrict__ Wq, const float* __restrict__ Wk,
    const float* __restrict__ Wv,
    float* __restrict__ q, float* __restrict__ k, float* __restrict__ v) {
    __shared__ float sW[3 * D_MODEL * D_MODEL];   // 48 KB
    __shared__ float sX[16 * D_MODEL];            // 4 KB

    const int rowBase = blockIdx.x * 16;

#if HAVE_TDM
    if (threadIdx.x < 32) {   // wave 0 drives the TDM
        unsigned base = (unsigned)(unsigned long long)(uintptr_t)&sW[0];
        tdm_load_mat64x64(Wq, base);
        tdm_load_mat64x64(Wk, base + 16384u);
        tdm_load_mat64x64(Wv, base + 32768u);
        __builtin_amdgcn_s_wait_tensorcnt(0);
    }
#else
    for (int i = threadIdx.x; i < D_MODEL * D_MODEL; i += 128) {
        sW[i]            = Wq[i];
        sW[4096 + i]     = Wk[i];
        sW[2 * 4096 + i] = Wv[i];
    }
#endif

    // stage x tile: 1024 floats = 256 float4, 2 per thread, coalesced
    {
        const float4* x4 = (const float4*)(x + (size_t)rowBase * D_MODEL);
        float4* sX4 = (float4*)sX;
        sX4[threadIdx.x]       = x4[threadIdx.x];
        sX4[threadIdx.x + 128] = x4[threadIdx.x + 128];
    }
    __syncthreads();

    const int lane    = threadIdx.x & 31;
    const int wave    = threadIdx.x >> 5;
    const int colBase = wave * 16;
    const int m15     = lane & 15;
    const int kHi     = (lane >> 4) * 2;   // lanes 16-31 take K+2,K+3

    // A fragments (shared by q/k/v): 32-bit A 16x4 layout,
    // lane(0-15)->K={kk,kk+1}, lane(16-31)->K={kk+2,kk+3}
    v2f aF[16];
    #pragma unroll
    for (int t = 0; t < 16; ++t) {
        aF[t].x = sX[m15 * D_MODEL + 4 * t + kHi];
        aF[t].y = sX[m15 * D_MODEL + 4 * t + kHi + 1];
    }

    for (int m = 0; m < 3; ++m) {
        const float* W = sW + m * 4096;
        v8f c = {};
        #pragma unroll
        for (int t = 0; t < 16; ++t) {
            const int kk = 4 * t;
            v2f b;   // 32-bit B 4x16 (mirrored striping): row K, col
            b.x = W[(kk + kHi)     * D_MODEL + colBase + m15];
            b.y = W[(kk + kHi + 1) * D_MODEL + colBase + m15];
            c = __builtin_amdgcn_wmma_f32_16x16x4_f32(
                    false, aF[t], false, b, (short)0, c, false, false);
        }
        float* O = (m == 0) ? q : (m == 1) ? k : v;
        // C/D layout: VGPR r -> M = r + 8*(lane>>4), N = lane&15
        const int orow = rowBase + (lane >> 4) * 8;
        const int ocol = colBase + m15;
        #pragma unroll
        for (int rr = 0; rr < 8; ++rr)
            O[(orow + rr) * D_MODEL + ocol] = c[rr];
    }
}

// ====================================================================
// Kernel 2: per-edge scores for all 8 heads + segment max via
// ordered-uint atomicMax on smax[dst][h].
// ====================================================================
__global__ __launch_bounds__(256) void edge_scores_kernel(
    const float* __restrict__ q, const float* __restrict__ k,
    const int* __restrict__ ei, const float* __restrict__ ea,
    const float* __restrict__ We,
    float* __restrict__ scores, unsigned* __restrict__ smax) {
    __shared__ float sWe[EDGE_IN * NUM_HEADS];
    if (threadIdx.x < EDGE_IN * NUM_HEADS) sWe[threadIdx.x] = We[threadIdx.x];
    __syncthreads();

    const int e = blockIdx.x * 256 + threadIdx.x;
    if (e >= N_EDGES) return;
    const int src = ei[e];
    const int dst = ei[N_EDGES + e];

    float a[EDGE_IN];
    const float4* ea4 = (const float4*)(ea + (size_t)e * EDGE_IN);
    #pragma unroll
    for (int i = 0; i < 4; ++i) {
        float4 t = ea4[i];
        a[4 * i] = t.x; a[4 * i + 1] = t.y; a[4 * i + 2] = t.z; a[4 * i + 3] = t.w;
    }

    const float4* qp = (const float4*)(q + (size_t)dst * D_MODEL);
    const float4* kp = (const float4*)(k + (size_t)src * D_MODEL);
    const float inv_sqrt_dh = 0.35355339059327373f;  // 1/sqrt(8)

    #pragma unroll
    for (int h = 0; h < NUM_HEADS; ++h) {
        float4 q0 = qp[2 * h], q1 = qp[2 * h + 1];
        float4 k0 = kp[2 * h], k1 = kp[2 * h + 1];
        float s = q0.x * k0.x + q0.y * k0.y + q0.z * k0.z + q0.w * k0.w +
                  q1.x * k1.x + q1.y * k1.y + q1.z * k1.z + q1.w * k1.w;
        float b = 0.0f;
        #pragma unroll
        for (int i = 0; i < EDGE_IN; ++i) b += a[i] * sWe[i * NUM_HEADS + h];
        s = s * inv_sqrt_dh + b;
        scores[(size_t)e * NUM_HEADS + h] = s;
        atomicMax(&smax[(size_t)dst * NUM_HEADS + h], f2ord(s));
    }
}

// ====================================================================
// Kernel 3: ex = exp(s - smax[dst]); overwrite scores with ex; atomicAdd
// denominator. One thread per (edge, head).
// ====================================================================
__global__ __launch_bounds__(256) void edge_exp_kernel(
    const int* __restrict__ ei, float* __restrict__ scores,
    const unsigned* __restrict__ smax, float* __restrict__ denom) {
    const size_t idx = (size_t)blockIdx.x * 256 + threadIdx.x;
    if (idx >= (size_t)N_EDGES * NUM_HEADS) return;
    const int e = (int)(idx >> 3);
    const int h = (int)(idx & 7);
    const int dst = ei[N_EDGES + e];
    const float mx = ord2f(smax[(size_t)dst * NUM_HEADS + h]);
    const float ex = __expf(scores[idx] - mx);
    scores[idx] = ex;
    atomicAdd(&denom[(size_t)dst * NUM_HEADS + h], ex);
}

// ====================================================================
// Kernel 4: alpha-weighted aggregation. One thread per (edge, head):
// agg[dst][h*8+d] += alpha * (v[src][h*8+d] + (edge_vec @ Wev)[h*8+d])
// ====================================================================
__global__ __launch_bounds__(256) void edge_agg_kernel(
    const int* __restrict__ ei, const float* __restrict__ v,
    const float* __restrict__ evec, const float* __restrict__ Wev,
    const float* __restrict__ scores, const float* __restrict__ denom,
    float* __restrict__ agg) {
    __shared__ float sWev[3 * D_MODEL];
    if (threadIdx.x < 3 * D_MODEL) sWev[threadIdx.x] = Wev[threadIdx.x];
    __syncthreads();

    const size_t idx = (size_t)blockIdx.x * 256 + threadIdx.x;
    if (idx >= (size_t)N_EDGES * NUM_HEADS) return;
    const int e = (int)(idx >> 3);
    const int h = (int)(idx & 7);
    const int src = ei[e];
    const int dst = ei[N_EDGES + e];

    const float alpha =
        scores[idx] / (denom[(size_t)dst * NUM_HEADS + h] + SMAX_EPS);
    const float e0 = evec[(size_t)e * 3];
    const float e1 = evec[(size_t)e * 3 + 1];
    const float e2 = evec[(size_t)e * 3 + 2];

    const float* vp = v + (size_t)src * D_MODEL + h * D_HEAD;
    float* ap = agg + (size_t)dst * D_MODEL + h * D_HEAD;
    #pragma unroll
    for (int d = 0; d < D_HEAD; ++d) {
        const int c = h * D_HEAD + d;
        const float ve = e0 * sWev[c] + e1 * sWev[D_MODEL + c] +
                         e2 * sWev[2 * D_MODEL + c];
        atomicAdd(&ap[d], alpha * (vp[d] + ve));
    }
}

// ====================================================================
// Kernel 5: LayerNorm + residual. One wave32 per node (2 elems/lane),
// wave allreduce via __shfl_xor.
// ====================================================================
__global__ __launch_bounds__(256) void ln_residual_kernel(
    const float* __restrict__ x, const float* __restrict__ agg,
    const float* __restrict__ gamma, const float* __restrict__ beta,
    float* __restrict__ out) {
    const int node = blockIdx.x * 8 + (threadIdx.x >> 5);
    if (node >= N_NODES) return;
    const int lane = threadIdx.x & 31;

    const float y0 = agg[(size_t)node * D_MODEL + lane];
    const float y1 = agg[(size_t)node * D_MODEL + 32 + lane];
    float s = y0 + y1;
    #pragma unroll
    for (int o = 16; o > 0; o >>= 1) s += __shfl_xor(s, o, 32);
    const float mu = s * (1.0f / D_MODEL);
    const float d0 = y0 - mu, d1 = y1 - mu;
    float vs = d0 * d0 + d1 * d1;
    #pragma unroll
    for (int o = 16; o > 0; o >>= 1) vs += __shfl_xor(vs, o, 32);
    const float rstd = rsqrtf(vs * (1.0f / D_MODEL) + LN_EPS);

    out[(size_t)node * D_MODEL + lane] =
        x[(size_t)node * D_MODEL + lane] + d0 * rstd * gamma[lane] + beta[lane];
    out[(size_t)node * D_MODEL + 32 + lane] =
        x[(size_t)node * D_MODEL + 32 + lane] + d1 * rstd * gamma[32 + lane] +
        beta[32 + lane];
}

// ====================================================================
extern "C" void kernel_launch(void* const* d_in, const int* in_sizes, int n_in,
                              void* d_out, int out_size, void* d_ws,
                              size_t ws_size, hipStream_t stream) {
    const float* x    = (const float*)d_in[0];
    const int*   ei   = (const int*)d_in[1];     // edge_index [2, E]
    const float* ea   = (const float*)d_in[2];   // edge_attr  [E, 16]
    const float* evec = (const float*)d_in[3];   // edge_vec   [E, 3]
    // d_in[4] = num_nodes scalar (compile-time constant here)
    const float* Wq   = (const float*)d_in[5];
    const float* Wk   = (const float*)d_in[6];
    const float* Wv   = (const float*)d_in[7];
    const float* We   = (const float*)d_in[8];
    const float* Wev  = (const float*)d_in[9];
    const float* ln_g = (const float*)d_in[10];
    const float* ln_b = (const float*)d_in[11];
    float* out = (float*)d_out;

    // ---- workspace carve-up (all L2-resident: ~80 MB total) ----
    float*    q      = (float*)d_ws;                    // N*64
    float*    k      = q + (size_t)N_NODES * D_MODEL;   // N*64
    float*    v      = k + (size_t)N_NODES * D_MODEL;   // N*64
    float*    agg    = v + (size_t)N_NODES * D_MODEL;   // N*64
    unsigned* smax   = (unsigned*)(agg + (size_t)N_NODES * D_MODEL);  // N*8
    float*    denom  = (float*)(smax + (size_t)N_NODES * NUM_HEADS);  // N*8
    float*    scores = denom + (size_t)N_NODES * NUM_HEADS;           // E*8

    init_kernel<<<(N_NODES * D_MODEL + 255) / 256, 256, 0, stream>>>(
        agg, denom, smax);
    qkv_wmma_kernel<<<N_NODES / 16, 128, 0, stream>>>(x, Wq, Wk, Wv, q, k, v);
    edge_scores_kernel<<<(N_EDGES + 255) / 256, 256, 0, stream>>>(
        q, k, ei, ea, We, scores, smax);
    edge_exp_kernel<<<(N_EDGES * NUM_HEADS + 255) / 256, 256, 0, stream>>>(
        ei, scores, smax, denom);
    edge_agg_kernel<<<(N_EDGES * NUM_HEADS + 255) / 256, 256, 0, stream>>>(
        ei, v, evec, Wev, scores, denom, agg);
    ln_residual_kernel<<<(N_NODES + 7) / 8, 256, 0, stream>>>(
        x, agg, ln_g, ln_b, out);
}